// DWT_22213570855097
// MI455X (gfx1250) — compile-verified
//
#include <hip/hip_runtime.h>
#include <stdint.h>

// ---------------------------------------------------------------------------
// Haar DWT (single level, 2x2 polyphase combine) for x = (16,64,256,256) f32.
// out = [ LL (16,64,128,128) | highs (16,64,3,128,128) ] concatenated flat.
//
// MI455X strategy: pure streaming kernel (512 MiB traffic, ~23 us HBM floor).
// Stage 32x256 f32 tiles into LDS with the CDNA5 Tensor Data Mover
// (tensor_load_to_lds + s_wait_tensorcnt), combine in VALU, write back with
// coalesced non-temporal b128 stores.
// ---------------------------------------------------------------------------

typedef float    f32x4 __attribute__((ext_vector_type(4)));
typedef uint32_t u32x4 __attribute__((ext_vector_type(4)));
typedef int      i32x4 __attribute__((ext_vector_type(4)));
typedef int      i32x8 __attribute__((ext_vector_type(8)));

#define TILE_ROWS 32       // input rows per workgroup tile
#define IN_W      256      // input width
#define OUT_W     128      // output width (and height)
#define PLANES    (16 * 64)
#define TILES_PER_PLANE (256 / TILE_ROWS)   // 8

__global__ __launch_bounds__(256) void haar_dwt_tdm_kernel(
    const float* __restrict__ x,
    const float* __restrict__ weights,
    float* __restrict__ out)
{
    __shared__ __align__(16) float tile[TILE_ROWS * IN_W];   // 32 KiB

    const int tid   = threadIdx.x;
    const int plane = blockIdx.x >> 3;       // (b*64 + c)
    const int ti    = blockIdx.x & 7;        // tile-row index within the plane

    // Global element base of this tile: plane stride 256*256, tile = 32 rows.
    const float* gp = x + (size_t)plane * (IN_W * IN_W)
                        + (size_t)ti * TILE_ROWS * IN_W;

#if __has_builtin(__builtin_amdgcn_tensor_load_to_lds)
    // ---- CDNA5 Tensor Data Mover: DMA the 256x32 f32 tile into LDS ----
    if (tid == 0) {
        const uint64_t gaddr = (uint64_t)(uintptr_t)gp;

        // D# group 0 (128b): count=1 | lds_addr | global_addr[56:0] | type=2
        u32x4 g0;
        g0[0] = 1u;                                     // count=1 (valid)
        g0[1] = 0u;                                     // lds_addr: `tile` is the
                                                        // only static LDS object -> offset 0
        g0[2] = (uint32_t)(gaddr & 0xffffffffu);        // global_addr[31:0]
        g0[3] = (uint32_t)((gaddr >> 32) & 0x01ffffffu) // global_addr[56:32]
              | (2u << 30);                             // type=2 ("image")

        // D# group 1 (256b): data_size=4B, 2D tensor 256 x 32, tile 256 x 32,
        // row stride 256 elements. No multicast / barrier / pad / iterate.
        i32x8 g1;
        g1[0] = (2 << 16);                    // [17:16] data_size = 2 -> 4 bytes
        g1[1] = (IN_W & 0xffff) << 16;        // [63:48]  tensor_dim0 lo = 256
        g1[2] = (TILE_ROWS & 0xffff) << 16;   // [95:80]  tensor_dim1 lo = 32
        g1[3] = (IN_W & 0xffff) << 16;        // [127:112] tile_dim0 = 256
        g1[4] = TILE_ROWS;                    // [143:128] tile_dim1 = 32 (tile_dim2 = 0)
        g1[5] = IN_W;                         // [191:160] tensor_dim0_stride lo32 = 256
        g1[6] = 0;                            // stride hi / tensor_dim1_stride lo
        g1[7] = 0;

        i32x4 gz = {0, 0, 0, 0};              // groups 2/3 unused (2D tensor)

#if __has_include(<hip/amd_detail/amd_gfx1250_TDM.h>)
        // amdgpu-toolchain (clang-23 / therock-10.0): 6-arg form
        i32x8 gz8 = {0, 0, 0, 0, 0, 0, 0, 0};
        __builtin_amdgcn_tensor_load_to_lds(g0, g1, gz, gz, gz8, /*cpol=*/0);
#else
        // ROCm 7.2 (clang-22): 5-arg form
        __builtin_amdgcn_tensor_load_to_lds(g0, g1, gz, gz, /*cpol=*/0);
#endif
        __builtin_amdgcn_s_wait_tensorcnt(0);
    }
    __syncthreads();
    // The TDM builtin writes `tile` behind the compiler's back (it only sees
    // the integer LDS offset, not a pointer). Escape the pointer and declare
    // a memory clobber so the ds_load/VALU combine below is not折 folded to
    // undef / dead-code eliminated. Costs zero instructions.
    asm volatile("" : : "r"(&tile[0]) : "memory");
#else
    // ---- Fallback: cooperative b128 loads into LDS ----
    for (int i = tid; i < TILE_ROWS * IN_W / 4; i += 256) {
        ((f32x4*)tile)[i] = ((const f32x4*)gp)[i];
    }
    __syncthreads();
#endif

    // ---- Polyphase Haar combine (matches reference op order exactly) ----
    const float w  = weights[0];
    const float w2 = w * w;

    const int j  = tid & 31;     // 4 output columns: [4j, 4j+4)
    const int r0 = tid >> 5;     // output row group 0..7

    float* d_ll = out;
    float* d_hi = out + (size_t)PLANES * OUT_W * OUT_W;   // highs base

    #pragma unroll
    for (int p = 0; p < 2; ++p) {
        const int r = r0 + 8 * p;                 // output row within tile 0..15
        const f32x4* row0 = (const f32x4*)&tile[(2 * r)     * IN_W + 8 * j];
        const f32x4* row1 = (const f32x4*)&tile[(2 * r + 1) * IN_W + 8 * j];
        const f32x4 t0 = row0[0];   // a0 b0 a1 b1
        const f32x4 t1 = row0[1];   // a2 b2 a3 b3
        const f32x4 t2 = row1[0];   // c0 d0 c1 d1
        const f32x4 t3 = row1[1];   // c2 d2 c3 d3

        f32x4 ll4, lh4, hl4, hh4;
        #pragma unroll
        for (int i = 0; i < 4; ++i) {
            const float a = (i < 2) ? t0[2 * i]     : t1[2 * (i - 2)];
            const float b = (i < 2) ? t0[2 * i + 1] : t1[2 * (i - 2) + 1];
            const float c = (i < 2) ? t2[2 * i]     : t3[2 * (i - 2)];
            const float d = (i < 2) ? t2[2 * i + 1] : t3[2 * (i - 2) + 1];

            const float sab = a + b;
            const float scd = c + d;
            const float sbd = b + d;
            const float sac = a + c;
            const float ddc = d - c;
            const float dba = b - a;

            ll4[i] = w2 * (sab + scd);   // w2*(a+b+(c+d))
            lh4[i] = w2 * (scd - sab);   // w2*(c+d-(a+b))
            hl4[i] = w2 * (sbd - sac);   // w2*(b+d-(a+c))
            hh4[i] = w2 * (ddc - dba);   // w2*(d-c-(b-a))
        }

        const int orow = ti * (TILE_ROWS / 2) + r;          // output row 0..127
        const size_t ob = (size_t)plane * (OUT_W * OUT_W)
                        + (size_t)orow * OUT_W + 4 * j;
        const size_t hb = (size_t)plane * (3 * OUT_W * OUT_W)
                        + (size_t)orow * OUT_W + 4 * j;

        // Streaming data: non-temporal, fully coalesced 512B/wave b128 stores.
        __builtin_nontemporal_store(ll4, (f32x4*)(d_ll + ob));
        __builtin_nontemporal_store(lh4, (f32x4*)(d_hi + hb));
        __builtin_nontemporal_store(hl4, (f32x4*)(d_hi + hb + OUT_W * OUT_W));
        __builtin_nontemporal_store(hh4, (f32x4*)(d_hi + hb + 2 * OUT_W * OUT_W));
    }
}

extern "C" void kernel_launch(void* const* d_in, const int* in_sizes, int n_in,
                              void* d_out, int out_size, void* d_ws, size_t ws_size,
                              hipStream_t stream) {
    const float* x   = (const float*)d_in[0];
    const float* wts = (const float*)d_in[1];
    float* out       = (float*)d_out;

    (void)in_sizes; (void)n_in; (void)out_size; (void)d_ws; (void)ws_size;

    dim3 grid(PLANES * TILES_PER_PLANE);   // 1024 planes * 8 tiles = 8192 blocks
    dim3 block(256);                       // 8 wave32 per block
    haar_dwt_tdm_kernel<<<grid, block, 0, stream>>>(x, wts, out);
}